// MultiCellLSTM_61950608277588
// MI455X (gfx1250) — compile-verified
//
#include <hip/hip_runtime.h>

// ---------------------------------------------------------------------------
// MultiCellLSTM scan on gfx1250 (MI455X, wave32, WMMA).
//   B=16384, T=2048 (1024 pairs: cellB on x[:,2k,:1] then cellA on x[:,2k+1,:]), H=10.
//   gates = [h, x, 1] @ [Whh ; Wih ; (bih+bhh)]^T  via V_WMMA_F32_16X16X4_F32
//   K padded to 12 (cellB) / 13->16 (cellA); 40 gates padded to 3 N-tiles of 16.
// One wave owns 32 batch rows (lane == batch). Weights live in VGPRs.
// ---------------------------------------------------------------------------

typedef __attribute__((ext_vector_type(2))) float v2f;
typedef __attribute__((ext_vector_type(8))) float v8f;

#define HID 10
#define NG 40
#define TLEN 2048
#define PAIRS (TLEN / 2)
#define ROWSTRIDE 52                 // LDS row stride (floats): 16B-aligned rows
#define WAVES_PER_BLOCK 2
#define THREADS (WAVES_PER_BLOCK * 32)

__device__ __forceinline__ float swapx16(float x) {
  // ds_swizzle group-of-32, xor_mask=0x10: exchange lane halves
  return __int_as_float(__builtin_amdgcn_ds_swizzle(__float_as_int(x), 0x401F));
}
__device__ __forceinline__ float sigm(float x) {
  return __builtin_amdgcn_rcpf(1.0f + __expf(-x));
}
__device__ __forceinline__ float tanhfast(float x) {
  return 2.0f * sigm(2.0f * x) - 1.0f;
}

// One element of the packed B-matrix [Whh ; Wih ; bias ; 0] at (K row, gate n)
__device__ __forceinline__ float wElem(const float* Whh, const float* Wih, int nx,
                                       const float* bih, const float* bhh,
                                       int K, int n) {
  if (n >= NG) return 0.0f;
  if (K < HID) return Whh[n * HID + K];          // Whh is [40, H]
  if (K < HID + nx) return Wih[n * nx + (K - HID)];
  if (K == HID + nx) return bih[n] + bhh[n];
  return 0.0f;
}

// One LSTM cell step for 32 batch rows held lane-per-batch in (h, c).
// NCH = number of K=4 chunks (3 for cellB, 4 for cellA).
template <int NCH>
__device__ __forceinline__ void lstmCellStep(const float (&ex)[16],
                                             const v2f (&WB)[3][NCH],
                                             float* wlds, int lane, int nInT,
                                             bool hi, float (&h)[HID], float (&c)[HID]) {
  // ---- Build A matrices (16x4 f32) for both 16-row M-groups --------------
  // ISA A layout: lanes 0-15 -> M rows, VGPR v: K = 4q+v (lanes<16), 4q+v+2 (lanes>=16)
  v2f A[2][NCH];
#pragma unroll
  for (int q = 0; q < NCH; ++q) {
#pragma unroll
    for (int v = 0; v < 2; ++v) {
      float lo = ex[q * 4 + v];
      float hv = ex[q * 4 + v + 2];
      float sLo = swapx16(lo);   // lane l <-> l^16
      float sHv = swapx16(hv);
      A[0][q][v] = hi ? sHv : lo;  // group 0: batches = lanes 0-15
      A[1][q][v] = hi ? hv : sLo;  // group 1: batches = lanes 16-31
    }
  }
  // ---- GEMM: D[g][tile] = sum_q A[g][q] x WB[tile][q] --------------------
#pragma unroll
  for (int g = 0; g < 2; ++g) {
#pragma unroll
    for (int t = 0; t < 3; ++t) {
      v8f acc = {};
#pragma unroll
      for (int q = 0; q < NCH; ++q) {
        acc = __builtin_amdgcn_wmma_f32_16x16x4_f32(
            /*neg_a=*/false, A[g][q], /*neg_b=*/false, WB[t][q],
            /*c_mod=*/(short)0, acc, /*reuse_a=*/false, /*reuse_b=*/false);
      }
      // D layout: lane -> gate (N), VGPR r (+8 for hi half) -> batch row (M).
      // Transpose to gates[batch][gate] through per-wave LDS tile.
      int gate = t * 16 + nInT;
      if (gate < NG) {
#pragma unroll
        for (int r = 0; r < 8; ++r) {
          int row = g * 16 + r + (hi ? 8 : 0);
          wlds[row * ROWSTRIDE + gate] = acc[r];
        }
      }
    }
  }
  asm volatile("s_wait_dscnt 0" ::: "memory");  // DS is in-order per wave; belt+braces
  // ---- Gather this lane's 40 gates, apply activations --------------------
  float G[NG];
  const float4* rowp = (const float4*)&wlds[lane * ROWSTRIDE];
#pragma unroll
  for (int t = 0; t < NG / 4; ++t) {
    float4 v = rowp[t];
    G[4 * t + 0] = v.x; G[4 * t + 1] = v.y;
    G[4 * t + 2] = v.z; G[4 * t + 3] = v.w;
  }
#pragma unroll
  for (int j = 0; j < HID; ++j) {
    float ig = sigm(G[j]);
    float fg = sigm(G[HID + j]);
    float gg = tanhfast(G[2 * HID + j]);
    float og = sigm(G[3 * HID + j]);
    float cn = fmaf(fg, c[j], ig * gg);
    c[j] = cn;
    h[j] = og * tanhfast(cn);
  }
}

__global__ void __launch_bounds__(THREADS)
MultiCellLSTM_kernel(const float* __restrict__ x,
                     const float* __restrict__ WihA, const float* __restrict__ WhhA,
                     const float* __restrict__ bihA, const float* __restrict__ bhhA,
                     const float* __restrict__ WihB, const float* __restrict__ WhhB,
                     const float* __restrict__ bihB, const float* __restrict__ bhhB,
                     const float* __restrict__ Wlin, const float* __restrict__ blin,
                     float* __restrict__ out) {
  __shared__ float lds[WAVES_PER_BLOCK * 32 * ROWSTRIDE];

  const int lane = threadIdx.x & 31;
  const int waveId = threadIdx.x >> 5;
  const int nInT = lane & 15;
  const bool hi = lane >= 16;
  const int batch = blockIdx.x * THREADS + threadIdx.x;  // lane == batch-in-wave
  float* wlds = &lds[waveId * 32 * ROWSTRIDE];

  // ---- Preload loop-invariant B matrices (K x 16-gates chunks) -----------
  // B layout mirrors A: lane -> N (gate), VGPR v: K = 4q+v (+2 for hi half)
  v2f WBB[3][3];  // cellB: K rows = [Whh(10) ; WihB(1) ; bias] -> 3 chunks
  v2f WBA[3][4];  // cellA: K rows = [Whh(10) ; WihA(2) ; bias ; 0] -> 4 chunks
#pragma unroll
  for (int t = 0; t < 3; ++t) {
#pragma unroll
    for (int q = 0; q < 4; ++q) {
      int n = t * 16 + nInT;
      int kb = q * 4 + (hi ? 2 : 0);
      if (q < 3) {
        WBB[t][q][0] = wElem(WhhB, WihB, 1, bihB, bhhB, kb + 0, n);
        WBB[t][q][1] = wElem(WhhB, WihB, 1, bihB, bhhB, kb + 1, n);
      }
      WBA[t][q][0] = wElem(WhhA, WihA, 2, bihA, bhhA, kb + 0, n);
      WBA[t][q][1] = wElem(WhhA, WihA, 2, bihA, bhhA, kb + 1, n);
    }
  }
  float wl[HID];
#pragma unroll
  for (int j = 0; j < HID; ++j) wl[j] = Wlin[j];
  const float bl = blin[0];

  float h[HID], c[HID];
#pragma unroll
  for (int j = 0; j < HID; ++j) { h[j] = 0.0f; c[j] = 0.0f; }

  // x is [B, T, 2]: one 16-byte load per pair-step per lane (its own batch row)
  const float4* xb = (const float4*)(x + (size_t)batch * (TLEN * 2));

#pragma unroll 1
  for (int k = 0; k < PAIRS; ++k) {
    float4 xv = xb[k];
    __builtin_prefetch(&xb[k + 8], 0, 0);  // global_prefetch_b8, stream ahead

    float ex[16];
    // cellB on x[:, 2k, :1]: ex = [h0..h9, xB, 1, 0...]
#pragma unroll
    for (int j = 0; j < HID; ++j) ex[j] = h[j];
    ex[10] = xv.x; ex[11] = 1.0f;
    ex[12] = 0.0f; ex[13] = 0.0f; ex[14] = 0.0f; ex[15] = 0.0f;
    lstmCellStep<3>(ex, WBB, wlds, lane, nInT, hi, h, c);

    // cellA on x[:, 2k+1, :]: ex = [h0..h9, x0, x1, 1, 0...]
#pragma unroll
    for (int j = 0; j < HID; ++j) ex[j] = h[j];
    ex[10] = xv.z; ex[11] = xv.w; ex[12] = 1.0f;
    ex[13] = 0.0f; ex[14] = 0.0f; ex[15] = 0.0f;
    lstmCellStep<4>(ex, WBA, wlds, lane, nInT, hi, h, c);
  }

  // ---- Final linear + sigmoid -------------------------------------------
  float acc = bl;
#pragma unroll
  for (int j = 0; j < HID; ++j) acc = fmaf(h[j], wl[j], acc);
  out[batch] = sigm(acc);
}

extern "C" void kernel_launch(void* const* d_in, const int* in_sizes, int n_in,
                              void* d_out, int out_size, void* d_ws, size_t ws_size,
                              hipStream_t stream) {
  const float* x    = (const float*)d_in[0];
  const float* WihA = (const float*)d_in[1];
  const float* WhhA = (const float*)d_in[2];
  const float* bihA = (const float*)d_in[3];
  const float* bhhA = (const float*)d_in[4];
  const float* WihB = (const float*)d_in[5];
  const float* WhhB = (const float*)d_in[6];
  const float* bihB = (const float*)d_in[7];
  const float* bhhB = (const float*)d_in[8];
  const float* Wlin = (const float*)d_in[9];
  const float* blin = (const float*)d_in[10];
  float* out = (float*)d_out;

  const int B = in_sizes[0] / (TLEN * 2);  // 16384
  dim3 grid(B / THREADS), block(THREADS);
  MultiCellLSTM_kernel<<<grid, block, 0, stream>>>(
      x, WihA, WhhA, bihA, bhhA, WihB, WhhB, bihB, bhhB, Wlin, blin, out);
}